// my_GCN_81174881894904
// MI455X (gfx1250) — compile-verified
//
#include <hip/hip_runtime.h>
#include <stdint.h>

typedef __attribute__((ext_vector_type(2))) float v2f;
typedef __attribute__((ext_vector_type(4))) float v4f;
typedef __attribute__((ext_vector_type(8))) float v8f;

#define NNODES 100000
#define NEDGES 1600000

// ---------------------------------------------------------------------------
// Degree / normalization: deg[i] = 1 (self loop) + #incoming edges at col;
// dis[i] = rsqrt(deg[i]).  deg stored as u32, converted in place to f32.
// ---------------------------------------------------------------------------
__global__ void k_deg_init(uint32_t* __restrict__ deg, int n) {
    int i = blockIdx.x * blockDim.x + threadIdx.x;
    if (i < n) deg[i] = 1u;
}

__global__ void k_deg_count(const int* __restrict__ col, uint32_t* __restrict__ deg, int e) {
    int i = blockIdx.x * blockDim.x + threadIdx.x;
    if (i < e) atomicAdd(&deg[col[i]], 1u);
}

__global__ void k_dis(float* __restrict__ dis, int n) {
    int i = blockIdx.x * blockDim.x + threadIdx.x;
    if (i < n) {
        uint32_t d = ((const uint32_t*)dis)[i];
        dis[i] = rsqrtf((float)d);
    }
}

// ---------------------------------------------------------------------------
// T = (RELU ? max(X,0) : X) @ W   via V_WMMA_F32_16X16X4_F32 (exact f32).
// One wave handles a 16-row slab x full N (NT 16-wide tiles). K in steps of 4.
//
// A 16x4 layout:  lane l (m = l&15) holds K = 2*(l>>4)   in a.x,
//                                         K = 2*(l>>4)+1 in a.y
// B 4x16 layout:  lane l (n = l&15) holds the same K pair for column n.
// C/D 16x16:      vgpr r, lane l -> row r + 8*(l>>4), col l&15.
// ---------------------------------------------------------------------------
template <int NT, int K, int N, bool RELU>
__global__ __launch_bounds__(256) void k_gemm(const float* __restrict__ X,
                                              const float* __restrict__ W,
                                              float* __restrict__ T,
                                              int mtiles) {
    const int lane = threadIdx.x & 31;
    const int wid  = blockIdx.x * (blockDim.x >> 5) + (threadIdx.x >> 5);
    if (wid >= mtiles) return;  // uniform per wave -> EXEC all-1 for WMMA

    const int mrow = wid * 16 + (lane & 15);
    const int koff = (lane >> 4) << 1;  // 0 or 2
    const float* __restrict__ Xrow = X + (size_t)mrow * K;

    v8f acc[NT];
#pragma unroll
    for (int j = 0; j < NT; ++j) acc[j] = (v8f){0.f, 0.f, 0.f, 0.f, 0.f, 0.f, 0.f, 0.f};

    for (int ks = 0; ks < K; ks += 4) {
        float a0 = Xrow[ks + koff];
        float a1 = Xrow[ks + koff + 1];
        if (RELU) { a0 = fmaxf(a0, 0.f); a1 = fmaxf(a1, 0.f); }
        v2f a;
        a.x = a0; a.y = a1;
#pragma unroll
        for (int j = 0; j < NT; ++j) {
            const int n = j * 16 + (lane & 15);
            v2f b;
            b.x = W[(size_t)(ks + koff) * N + n];
            b.y = W[(size_t)(ks + koff + 1) * N + n];
            acc[j] = __builtin_amdgcn_wmma_f32_16x16x4_f32(
                /*neg_a=*/false, a, /*neg_b=*/false, b,
                /*c_mod=*/(short)0, acc[j], /*reuse_a=*/false, /*reuse_b=*/false);
        }
    }

    const int mbase = wid * 16 + ((lane >> 4) << 3);
#pragma unroll
    for (int j = 0; j < NT; ++j) {
#pragma unroll
        for (int r = 0; r < 8; ++r) {
            T[(size_t)(mbase + r) * N + j * 16 + (lane & 15)] = acc[j][r];
        }
    }
}

// ---------------------------------------------------------------------------
// O[i][c] = bias[c] + T[i][c] * dis[i]^2   (self-loop message + bias)
// ---------------------------------------------------------------------------
template <int C, int LOGC>
__global__ __launch_bounds__(256) void k_init_out(const float* __restrict__ T,
                                                  const float* __restrict__ bias,
                                                  const float* __restrict__ dis,
                                                  float* __restrict__ O) {
    unsigned i = blockIdx.x * blockDim.x + threadIdx.x;
    if (i >= (unsigned)NNODES * C) return;
    unsigned node = i >> LOGC;
    unsigned c = i & (C - 1);
    float d = dis[node];
    O[i] = bias[c] + T[i] * (d * d);
}

// ---------------------------------------------------------------------------
// Edge scatter: one thread per (edge, 4-feature chunk).
//   O[col][c..c+3] += T[row][c..c+3] * dis[row]*dis[col]
// ---------------------------------------------------------------------------
template <int C, int LOGCH>
__global__ __launch_bounds__(256) void k_scatter(const int* __restrict__ row,
                                                 const int* __restrict__ col,
                                                 const float* __restrict__ dis,
                                                 const float* __restrict__ T,
                                                 float* __restrict__ O) {
    constexpr int CH = C / 4;
    unsigned tid = blockIdx.x * blockDim.x + threadIdx.x;
    if (tid >= (unsigned)NEDGES * CH) return;
    unsigned e  = tid >> LOGCH;
    unsigned ch = tid & (CH - 1);
    int r = row[e];
    int c = col[e];
    float nrm = dis[r] * dis[c];
    v4f v = *(const v4f*)(T + (size_t)r * C + ch * 4u);
    float* dst = O + (size_t)c * C + ch * 4u;
    __hip_atomic_fetch_add(dst + 0, v.x * nrm, __ATOMIC_RELAXED, __HIP_MEMORY_SCOPE_AGENT);
    __hip_atomic_fetch_add(dst + 1, v.y * nrm, __ATOMIC_RELAXED, __HIP_MEMORY_SCOPE_AGENT);
    __hip_atomic_fetch_add(dst + 2, v.z * nrm, __ATOMIC_RELAXED, __HIP_MEMORY_SCOPE_AGENT);
    __hip_atomic_fetch_add(dst + 3, v.w * nrm, __ATOMIC_RELAXED, __HIP_MEMORY_SCOPE_AGENT);
}

// ---------------------------------------------------------------------------
extern "C" void kernel_launch(void* const* d_in, const int* in_sizes, int n_in,
                              void* d_out, int out_size, void* d_ws, size_t ws_size,
                              hipStream_t stream) {
    const float* x   = (const float*)d_in[0];
    const int*   ei  = (const int*)d_in[1];
    const float* W1  = (const float*)d_in[2];
    const float* b1  = (const float*)d_in[3];
    const float* W2  = (const float*)d_in[4];
    const float* b2  = (const float*)d_in[5];
    const float* W3  = (const float*)d_in[6];
    const float* b3  = (const float*)d_in[7];
    float*       out = (float*)d_out;

    const int* rowp = ei;            // sources
    const int* colp = ei + NEDGES;   // targets

    // Workspace layout (floats): [dis | bufA (100000x128) | bufB (100000x128)]
    float* ws   = (float*)d_ws;
    float* dis  = ws;                    // NNODES entries (u32 deg, then f32 dis)
    float* bufA = ws + 131072;           // 16-B aligned
    float* bufB = bufA + (size_t)NNODES * 128;

    const int MT = NNODES / 16;          // 6250 row tiles (exact)
    const int gemm_blocks = (MT + 7) / 8;  // 8 waves (256 thr) per block

    // ---- normalization ----
    k_deg_init<<<(NNODES + 255) / 256, 256, 0, stream>>>((uint32_t*)dis, NNODES);
    k_deg_count<<<(NEDGES + 255) / 256, 256, 0, stream>>>(colp, (uint32_t*)dis, NEDGES);
    k_dis<<<(NNODES + 255) / 256, 256, 0, stream>>>(dis, NNODES);

    // ---- layer 1: 256 -> 128 ----
    k_gemm<8, 256, 128, false><<<gemm_blocks, 256, 0, stream>>>(x, W1, bufA, MT);
    k_init_out<128, 7><<<((unsigned)NNODES * 128 + 255) / 256, 256, 0, stream>>>(bufA, b1, dis, bufB);
    k_scatter<128, 5><<<((unsigned)NEDGES * 32 + 255) / 256, 256, 0, stream>>>(rowp, colp, dis, bufA, bufB);

    // ---- layer 2: 128 -> 64 (ReLU folded into A-load) ----
    k_gemm<4, 128, 64, true><<<gemm_blocks, 256, 0, stream>>>(bufB, W2, bufA, MT);
    k_init_out<64, 6><<<((unsigned)NNODES * 64 + 255) / 256, 256, 0, stream>>>(bufA, b2, dis, bufB);
    k_scatter<64, 4><<<((unsigned)NEDGES * 16 + 255) / 256, 256, 0, stream>>>(rowp, colp, dis, bufA, bufB);

    // ---- layer 3: 64 -> 32, written straight to d_out ----
    k_gemm<2, 64, 32, true><<<gemm_blocks, 256, 0, stream>>>(bufB, W3, bufA, MT);
    k_init_out<32, 5><<<((unsigned)NNODES * 32 + 255) / 256, 256, 0, stream>>>(bufA, b3, dis, out);
    k_scatter<32, 3><<<((unsigned)NEDGES * 8 + 255) / 256, 256, 0, stream>>>(rowp, colp, dis, bufA, out);
}